// WaveletDenoise1D_39831526703793
// MI455X (gfx1250) — compile-verified
//
#include <hip/hip_runtime.h>
#include <hip/hip_bf16.h>
#include <math.h>

typedef __attribute__((ext_vector_type(2))) float v2f;
typedef __attribute__((ext_vector_type(8))) float v8f;

// Only meaningful in the device pass; the host pass has no amdgcn builtins.
#if defined(__HIP_DEVICE_COMPILE__)
#if !__has_builtin(__builtin_amdgcn_wmma_f32_16x16x4_f32)
#error "wmma f32 16x16x4 builtin not available on this toolchain (device pass)"
#endif
#endif

__device__ __forceinline__ v8f wmma4(v2f a, v2f b, v8f c) {
  // 8 args: (neg_a, A, neg_b, B, c_mod, C, reuse_a, reuse_b)
  return __builtin_amdgcn_wmma_f32_16x16x4_f32(false, a, false, b, (short)0, c,
                                               false, false);
}

// db4-style taps from the reference:
// analysis low  G_lo[t] = H[t]
// analysis high G_hi[t] = (-1)^t * H[7-t]  (== synthesis high REC_HI[t])
__device__ __forceinline__ float tapLo(int t) {
  float r = 0.0f;
  r = (t == 0) ? 0.23037781330885523f : r;
  r = (t == 1) ? 0.7148465705525415f : r;
  r = (t == 2) ? 0.6308807679295904f : r;
  r = (t == 3) ? -0.02798376941698385f : r;
  r = (t == 4) ? -0.18703481171888114f : r;
  r = (t == 5) ? 0.030841381835986965f : r;
  r = (t == 6) ? 0.032883011666982945f : r;
  r = (t == 7) ? -0.010597401784997278f : r;
  return r;
}
__device__ __forceinline__ float tapHi(int t) {
  float r = 0.0f;
  r = (t == 0) ? -0.010597401784997278f : r;
  r = (t == 1) ? -0.032883011666982945f : r;
  r = (t == 2) ? 0.030841381835986965f : r;
  r = (t == 3) ? 0.18703481171888114f : r;
  r = (t == 4) ? -0.02798376941698385f : r;
  r = (t == 5) ? -0.6308807679295904f : r;
  r = (t == 6) ? 0.7148465705525415f : r;
  r = (t == 7) ? -0.23037781330885523f : r;
  return r;
}

// ---------------------------------------------------------------------------
// DWT level: for each row r and output i:
//   a[i] = sum_t tapLo(t) * x[2i-6+t],  d[i] = sum_t tapHi(t) * x[2i-6+t]
// One wave per (16-row, 16-output) tile. Input window = 40 samples, staged in
// LDS; A(16x40) x Wband(40x16) via 10 chained V_WMMA_F32_16X16X4_F32 per
// filter. Wband[k][j] = tap(k - 2j) (banded, built from constants per lane).
// ---------------------------------------------------------------------------
#define LPITCH 42  // even (8B-aligned rows); 42*L mod 64 distinct -> no bank conflicts

__global__ __launch_bounds__(32) void dwt_wmma(const float* __restrict__ x, int N,
                                               float* __restrict__ aout,
                                               float* __restrict__ dout, int Nout) {
  __shared__ float tile[16 * LPITCH];
  const int lane = threadIdx.x;
  const int rowBase = blockIdx.y << 4;
  const int colBase = blockIdx.x << 4;
  const int base = 2 * colBase - 6;  // x index of window k=0 (may be negative)

  // Stage 16 rows x 40 window columns (zero-filled OOB) into LDS, coalesced.
#pragma unroll
  for (int it = 0; it < 20; ++it) {
    int idx = lane + (it << 5);  // 0..639
    int r = idx / 40;
    int cc = idx - r * 40;
    int gx = base + cc;
    float v = 0.0f;
    if (gx >= 0 && gx < N) v = x[(size_t)(rowBase + r) * (size_t)N + gx];
    tile[r * LPITCH + cc] = v;
  }
  __syncthreads();

  const int half = lane >> 4;  // which K-pair half this lane holds
  const int sub = lane & 15;   // A: matrix row M;  B/D: column N
  const int aoff = half * 2;
  const int j2 = 2 * sub;
  const float* trow = &tile[sub * LPITCH];

  v8f accLo = {0.f, 0.f, 0.f, 0.f, 0.f, 0.f, 0.f, 0.f};
  v8f accHi = {0.f, 0.f, 0.f, 0.f, 0.f, 0.f, 0.f, 0.f};
#pragma unroll
  for (int s = 0; s < 10; ++s) {
    const int k0 = 4 * s + aoff;  // K index for VGPR0 of this lane-half
    v2f a;
    a.x = trow[k0];
    a.y = trow[k0 + 1];
    const int t0 = k0 - j2;  // band offset
    v2f bl, bh;
    bl.x = tapLo(t0);
    bl.y = tapLo(t0 + 1);
    bh.x = tapHi(t0);
    bh.y = tapHi(t0 + 1);
    accLo = wmma4(a, bl, accLo);
    accHi = wmma4(a, bh, accHi);
  }

  const int ocol = colBase + sub;
  if (ocol < Nout) {
    const int r0 = rowBase + half * 8;
#pragma unroll
    for (int v = 0; v < 8; ++v) {
      size_t o = (size_t)(r0 + v) * (size_t)Nout + ocol;
      aout[o] = accLo[v];
      dout[o] = accHi[v];
    }
  }
}

// ---------------------------------------------------------------------------
// IDWT level: rec[i] = sum_j tapLo(i-2j+6)*a[j] + sum_j tapHi(i-2j+6)*d[j].
// 16-output tile needs j in [i0/2-1, i0/2+10]; use jbase=i0/2-2, K=16.
// Band: W[kk][n] = tap(n - 2kk + 10). 4 chained WMMAs per operand.
// Stores to out[row*opitch + (i - store_lo)] for store_lo <= i < Nout.
// ---------------------------------------------------------------------------
#define IPITCH 18

__global__ __launch_bounds__(32) void idwt_wmma(const float* __restrict__ a, int apitch,
                                                const float* __restrict__ d, int Na,
                                                float* __restrict__ out, int Nout,
                                                int opitch, int store_lo) {
  __shared__ float ta[16 * IPITCH];
  __shared__ float td[16 * IPITCH];
  const int lane = threadIdx.x;
  const int rowBase = blockIdx.y << 4;
  const int i0 = store_lo + (blockIdx.x << 4);
  const int jbase = (i0 >> 1) - 2;

  // Stage 16x16 a-tile and 16x16 d-tile (zero-filled OOB).
#pragma unroll
  for (int it = 0; it < 16; ++it) {
    int idx = lane + (it << 5);  // 0..511
    int which = idx >> 8;        // 0 -> a, 1 -> d
    int e = idx & 255;
    int r = e >> 4;
    int cc = e & 15;
    int gj = jbase + cc;
    float v = 0.0f;
    if (gj >= 0 && gj < Na) {
      const float* src = which ? d : a;
      int pitch = which ? Na : apitch;
      v = src[(size_t)(rowBase + r) * (size_t)pitch + gj];
    }
    float* dst = which ? td : ta;
    dst[r * IPITCH + cc] = v;
  }
  __syncthreads();

  const int half = lane >> 4;
  const int sub = lane & 15;
  const int aoff = half * 2;
  const float* ra = &ta[sub * IPITCH];
  const float* rd = &td[sub * IPITCH];

  v8f acc = {0.f, 0.f, 0.f, 0.f, 0.f, 0.f, 0.f, 0.f};
#pragma unroll
  for (int s = 0; s < 4; ++s) {
    const int k0 = 4 * s + aoff;
    const int t0 = sub - 2 * k0 + 10;
    v2f av, dv, bl, bh;
    av.x = ra[k0];
    av.y = ra[k0 + 1];
    dv.x = rd[k0];
    dv.y = rd[k0 + 1];
    bl.x = tapLo(t0);
    bl.y = tapLo(t0 - 2);
    bh.x = tapHi(t0);
    bh.y = tapHi(t0 - 2);
    acc = wmma4(av, bl, acc);
    acc = wmma4(dv, bh, acc);
  }

  const int i = i0 + sub;
  if (i < Nout) {
    const int r0 = rowBase + half * 8;
#pragma unroll
    for (int v = 0; v < 8; ++v) {
      out[(size_t)(r0 + v) * (size_t)opitch + (size_t)(i - store_lo)] = acc[v];
    }
  }
}

// ---------------------------------------------------------------------------
// Per-row RMS + soft-threshold shrinkage (in place on the detail buffer).
// sigma = max(sqrt(mean(d^2)), 1e-6); thr = softplus(theta)*sigma;
// out = rho*d + (1-rho)*alpha*sign(d)*relu(|d|-thr)
// ---------------------------------------------------------------------------
__global__ __launch_bounds__(256) void thresh_k(float* __restrict__ d, int Nd,
                                                const float* __restrict__ theta,
                                                const float* __restrict__ phi,
                                                const float* __restrict__ psi,
                                                int lvl, int C) {
  __shared__ float red[256];
  const int row = blockIdx.x;  // row = b*C + c
  const int c = row % C;
  float* dr = d + (size_t)row * (size_t)Nd;

  float s = 0.0f;
  for (int i = threadIdx.x; i < Nd; i += 256) {
    float v = dr[i];
    s += v * v;
  }
  red[threadIdx.x] = s;
  __syncthreads();
  for (int w = 128; w > 0; w >>= 1) {
    if (threadIdx.x < w) red[threadIdx.x] += red[threadIdx.x + w];
    __syncthreads();
  }
  const float sigma = fmaxf(sqrtf(red[0] / (float)Nd), 1e-6f);

  const float th = theta[lvl * C + c];
  const float ph = phi[lvl * C + c];
  const float ps = psi[lvl * C + c];
  const float tau = (th > 20.0f) ? th : log1pf(expf(th));
  const float alpha = 1.0f / (1.0f + expf(-ph));
  const float rho = 1.0f / (1.0f + expf(-ps));
  const float thr = tau * sigma;

  for (int i = threadIdx.x; i < Nd; i += 256) {
    float v = dr[i];
    float sh = copysignf(fmaxf(fabsf(v) - thr, 0.0f), v);
    dr[i] = rho * v + (1.0f - rho) * (alpha * sh);
  }
}

// ---------------------------------------------------------------------------
// Host-side orchestration.
// Sizes fixed by the reference: B=16, C=128, T=9216, level=4, target=8192.
// Level lengths: 9216 -> 4611 -> 2309 -> 1158 -> 582.
// IDWT outputs: 582->1158, 1158->2310 (trunc 2309), 2309->4612 (trunc 4611),
// 4611->9216 (keep last 8192 -> d_out).
// ---------------------------------------------------------------------------
extern "C" void kernel_launch(void* const* d_in, const int* in_sizes, int n_in,
                              void* d_out, int out_size, void* d_ws, size_t ws_size,
                              hipStream_t stream) {
  (void)in_sizes; (void)n_in; (void)out_size; (void)ws_size;
  const float* x = (const float*)d_in[0];
  const float* theta = (const float*)d_in[1];
  const float* phi = (const float*)d_in[2];
  const float* psi = (const float*)d_in[3];
  float* out = (float*)d_out;
  float* ws = (float*)d_ws;

  const int BC = 2048, C = 128;
  const int N0 = 9216, N1 = 4611, N2 = 2309, N3 = 1158, N4 = 582;
  const int TARGET = 8192;

  size_t off = 0;
  float* D1 = ws + off; off += (size_t)BC * N1;
  float* D2 = ws + off; off += (size_t)BC * N2;
  float* D3 = ws + off; off += (size_t)BC * N3;
  float* D4 = ws + off; off += (size_t)BC * N4;
  float* A1 = ws + off; off += (size_t)BC * (N1 + 1);  // reused for rec (4612)
  float* A2 = ws + off; off += (size_t)BC * (N2 + 1);  // reused for rec (2310)
  float* A3 = ws + off; off += (size_t)BC * N3;        // reused for rec (1158)
  float* A4 = ws + off; off += (size_t)BC * N4;

  const dim3 blk(32);
  const int RB = BC / 16;  // 128 row tiles
  auto ct = [](int n) { return (n + 15) / 16; };

  // Forward DWT cascade
  dwt_wmma<<<dim3(ct(N1), RB), blk, 0, stream>>>(x,  N0, A1, D1, N1);
  dwt_wmma<<<dim3(ct(N2), RB), blk, 0, stream>>>(A1, N1, A2, D2, N2);
  dwt_wmma<<<dim3(ct(N3), RB), blk, 0, stream>>>(A2, N2, A3, D3, N3);
  dwt_wmma<<<dim3(ct(N4), RB), blk, 0, stream>>>(A3, N3, A4, D4, N4);

  // Shrinkage (level index follows details[::-1]: 0->d4 ... 3->d1)
  thresh_k<<<BC, 256, 0, stream>>>(D4, N4, theta, phi, psi, 0, C);
  thresh_k<<<BC, 256, 0, stream>>>(D3, N3, theta, phi, psi, 1, C);
  thresh_k<<<BC, 256, 0, stream>>>(D2, N2, theta, phi, psi, 2, C);
  thresh_k<<<BC, 256, 0, stream>>>(D1, N1, theta, phi, psi, 3, C);

  // Inverse cascade (rec buffers reuse A3/A2/A1; reads never alias writes)
  idwt_wmma<<<dim3(ct(N3), RB), blk, 0, stream>>>(A4, N4, D4, N4, A3, N3, N3, 0);
  idwt_wmma<<<dim3(ct(2 * N3 - 6), RB), blk, 0, stream>>>(A3, N3, D3, N3, A2,
                                                          2 * N3 - 6, 2 * N3 - 6, 0);
  idwt_wmma<<<dim3(ct(2 * N2 - 6), RB), blk, 0, stream>>>(A2, 2 * N3 - 6, D2, N2, A1,
                                                          2 * N2 - 6, 2 * N2 - 6, 0);
  // Final level: only the last TARGET samples, straight into d_out.
  idwt_wmma<<<dim3(TARGET / 16, RB), blk, 0, stream>>>(
      A1, 2 * N2 - 6, D1, N1, out, N0, TARGET, N0 - TARGET);
}